// Adaptive_Spectral_Block_t_73005854098171
// MI455X (gfx1250) — compile-verified
//
#include <hip/hip_runtime.h>
#include <math.h>

#define B_    8
#define C_    128
#define N_    4096
#define F_    2049          // N/2+1
#define LOG2N 12
#define NGT   129           // ceil(F/16) g-tiles / f-tiles
#define GP    2064          // padded row length (NGT*16)

typedef float v2f __attribute__((ext_vector_type(2)));
typedef float v8f __attribute__((ext_vector_type(8)));

// ---------------------------------------------------------------------------
// fp32 WMMA helper: D = A(16x4) * B(4x16) + C  (exact fp32 path on CDNA5)
// A layout: lane L, comp j -> (M = L%16, K = 2*(L/16)+j)
// B layout: lane L, comp j -> (K = 2*(L/16)+j, N = L%16)
// C/D layout: vgpr r, lane L -> (M = r + 8*(L/16), N = L%16)
// ---------------------------------------------------------------------------
__device__ __forceinline__ v8f wmma_acc(v2f a, v2f b, v8f c) {
    return __builtin_amdgcn_wmma_f32_16x16x4_f32(false, a, false, b,
                                                 (short)0, c, false, false);
}

// ---------------------------------------------------------------------------
// Projection: xKt[b][c][n] = sum_k W_K[c][k]*x_in[b][k][n] + b_K[c]; same for V.
// One wave computes one 16x16 output tile (K=128 => 32 WMMA steps, 2 outputs).
// ---------------------------------------------------------------------------
__global__ __launch_bounds__(256) void proj_kernel(
    const float* __restrict__ x,                     // (B,C,N) = x_in
    const float* __restrict__ Wk, const float* __restrict__ bk,
    const float* __restrict__ Wv, const float* __restrict__ bv,
    float* __restrict__ xKt, float* __restrict__ xVt)
{
    const int wave = threadIdx.x >> 5;
    const int lane = threadIdx.x & 31;
    const int lm = lane & 15, lh = lane >> 4;
    const int tilesPerBatch = (C_ / 16) * (N_ / 16);           // 2048
    long tile = (long)blockIdx.x * 8 + wave;                   // 16384 tiles total
    const int b  = (int)(tile / tilesPerBatch);
    const int t  = (int)(tile % tilesPerBatch);
    const int m0 = (t / (N_ / 16)) * 16;                       // output-channel tile
    const int n0 = (t % (N_ / 16)) * 16;                       // time tile
    const float* xb = x + (size_t)b * C_ * N_;

    v8f accK = {}; v8f accV = {};
#pragma unroll 4
    for (int k0 = 0; k0 < C_; k0 += 4) {
        const int ka = k0 + 2 * lh;
        v2f aK = { Wk[(m0 + lm) * C_ + ka], Wk[(m0 + lm) * C_ + ka + 1] };
        v2f aV = { Wv[(m0 + lm) * C_ + ka], Wv[(m0 + lm) * C_ + ka + 1] };
        v2f bb = { xb[(size_t)ka * N_ + n0 + lm],
                   xb[(size_t)(ka + 1) * N_ + n0 + lm] };
        accK = wmma_acc(aK, bb, accK);
        accV = wmma_acc(aV, bb, accV);
    }
#pragma unroll
    for (int r = 0; r < 8; ++r) {
        const int M = m0 + r + 8 * lh;
        const size_t o = ((size_t)b * C_ + M) * N_ + n0 + lm;
        xKt[o] = accK[r] + bk[M];
        xVt[o] = accV[r] + bv[M];
    }
}

// ---------------------------------------------------------------------------
// Forward 4096-pt FFT (radix-2 DIT in LDS), real input, ortho norm (1/64).
// One (b,c) row per block. Writes split re/im planes in (B,F,C).
// ---------------------------------------------------------------------------
__global__ __launch_bounds__(256) void fft_fwd_kernel(
    const float* __restrict__ src,                   // (B,C,N)
    float* __restrict__ outRe, float* __restrict__ outIm)
{
    __shared__ float re[N_];
    __shared__ float im[N_];
    const int b = blockIdx.x / C_, c = blockIdx.x % C_;
    const float* row = src + ((size_t)b * C_ + c) * N_;

    for (int n = threadIdx.x; n < N_; n += blockDim.x) {
        const unsigned r = __brev((unsigned)n) >> (32 - LOG2N);
        re[r] = row[n];
        im[r] = 0.0f;
    }
    __syncthreads();

    for (int s = 1; s <= LOG2N; ++s) {
        const int m = 1 << s, half = m >> 1;
        for (int t = threadIdx.x; t < (N_ >> 1); t += blockDim.x) {
            const int j  = t & (half - 1);
            const int i0 = ((t >> (s - 1)) << s) + j;
            const int i1 = i0 + half;
            const float ang = -6.283185307179586f * (float)j / (float)m;
            float si, co; __sincosf(ang, &si, &co);
            const float xr = re[i1], xi = im[i1];
            const float tr = co * xr - si * xi;
            const float ti = co * xi + si * xr;
            const float ur = re[i0], ui = im[i0];
            re[i0] = ur + tr; im[i0] = ui + ti;
            re[i1] = ur - tr; im[i1] = ui - ti;
        }
        __syncthreads();
    }

    const float sc = 1.0f / 64.0f;                   // ortho: 1/sqrt(4096)
    for (int f = threadIdx.x; f < F_; f += blockDim.x) {
        const size_t o = ((size_t)b * F_ + f) * C_ + c;
        outRe[o] = re[f] * sc;
        outIm[o] = im[f] * sc;
    }
}

// ---------------------------------------------------------------------------
// Inverse: Hermitian-extend 2049 bins to 4096, inverse FFT, real part -> (B,C,N)
// ---------------------------------------------------------------------------
__global__ __launch_bounds__(256) void fft_inv_kernel(
    const float* __restrict__ Cre, const float* __restrict__ Cim,
    float* __restrict__ out)                         // (B,C,N)
{
    __shared__ float re[N_];
    __shared__ float im[N_];
    const int b = blockIdx.x / C_, c = blockIdx.x % C_;
    const float* pr = Cre + (size_t)b * F_ * C_ + c;
    const float* pi = Cim + (size_t)b * F_ * C_ + c;

    for (int n = threadIdx.x; n < N_; n += blockDim.x) {
        float vr, vi;
        if (n < F_) { vr = pr[(size_t)n * C_];          vi =  pi[(size_t)n * C_]; }
        else { const int k2 = N_ - n;
               vr = pr[(size_t)k2 * C_];                vi = -pi[(size_t)k2 * C_]; }
        const unsigned r = __brev((unsigned)n) >> (32 - LOG2N);
        re[r] = vr; im[r] = vi;
    }
    __syncthreads();

    for (int s = 1; s <= LOG2N; ++s) {
        const int m = 1 << s, half = m >> 1;
        for (int t = threadIdx.x; t < (N_ >> 1); t += blockDim.x) {
            const int j  = t & (half - 1);
            const int i0 = ((t >> (s - 1)) << s) + j;
            const int i1 = i0 + half;
            const float ang = 6.283185307179586f * (float)j / (float)m;
            float si, co; __sincosf(ang, &si, &co);
            const float xr = re[i1], xi = im[i1];
            const float tr = co * xr - si * xi;
            const float ti = co * xi + si * xr;
            const float ur = re[i0], ui = im[i0];
            re[i0] = ur + tr; im[i0] = ui + ti;
            re[i1] = ur - tr; im[i1] = ui - ti;
        }
        __syncthreads();
    }

    const float sc = 1.0f / 64.0f;
    float* orow = out + ((size_t)b * C_ + c) * N_;
    for (int n = threadIdx.x; n < N_; n += blockDim.x)
        orow[n] = re[n] * sc;
}

// ---------------------------------------------------------------------------
// energy[b][f] = sum_c |x_fft|^2 ; one wave per (b,f)
// ---------------------------------------------------------------------------
__global__ __launch_bounds__(32) void energy_kernel(
    const float* __restrict__ Xre, const float* __restrict__ Xim,
    float* __restrict__ energy)
{
    const int idx = blockIdx.x;                       // b*F + f
    const int lane = threadIdx.x;
    const float* pr = Xre + (size_t)idx * C_;
    const float* pi = Xim + (size_t)idx * C_;
    float s = 0.0f;
    for (int c = lane; c < C_; c += 32) {
        const float a = pr[c], q = pi[c];
        s += a * a + q * q;
    }
    for (int off = 16; off; off >>= 1) s += __shfl_xor(s, off);
    if (lane == 0) energy[idx] = s;
}

// ---------------------------------------------------------------------------
// Exact k-th order statistic by binary search on (non-negative) float bits.
// mode 0: vals = energy[bsel*F_ + i], i<n.  mode 1: vals = energy[i]/(med[i/F_]+1e-6).
// ---------------------------------------------------------------------------
__device__ float block_select_kth(const float* __restrict__ energy,
                                  const float* __restrict__ med,
                                  int mode, int bsel, int n, int k, int* scnt)
{
    unsigned lo = 0u, hi = 0x7F800000u;               // [0, +inf]
    while (lo < hi) {
        const unsigned mid = (lo + hi) >> 1;
        const float mv = __uint_as_float(mid);
        int local = 0;
        for (int i = threadIdx.x; i < n; i += blockDim.x) {
            float v;
            if (mode == 0) v = energy[bsel * F_ + i];
            else           v = energy[i] / (med[i / F_] + 1e-6f);
            if (v <= mv) ++local;
        }
        __syncthreads();
        if (threadIdx.x == 0) *scnt = 0;
        __syncthreads();
        atomicAdd(scnt, local);
        __syncthreads();
        const int cnt = *scnt;
        if (cnt >= k + 1) hi = mid; else lo = mid + 1;
    }
    return __uint_as_float(lo);
}

__global__ __launch_bounds__(256) void medquant_kernel(
    const float* __restrict__ energy, const float* __restrict__ thParam,
    float* __restrict__ stats)                        // stats[0]=threshold, [1..8]=medians
{
    __shared__ int   scnt;
    __shared__ float smed[B_];
    for (int b = 0; b < B_; ++b) {
        const float m = block_select_kth(energy, smed, 0, b, F_, (F_ - 1) / 2, &scnt);
        if (threadIdx.x == 0) smed[b] = m;
        __syncthreads();
    }
    const float q = thParam[0];
    const int n = B_ * F_;
    float pos = q * (float)(n - 1);
    if (pos < 0.0f) pos = 0.0f;
    const float r0f = floorf(pos);
    int r0 = (int)r0f; if (r0 > n - 1) r0 = n - 1;
    const float frac = pos - r0f;
    const float v0 = block_select_kth(energy, smed, 1, 0, n, r0, &scnt);
    const int r1 = (r0 + 1 < n) ? r0 + 1 : n - 1;
    const float v1 = block_select_kth(energy, smed, 1, 0, n, r1, &scnt);
    if (threadIdx.x == 0) {
        stats[0] = v0 + (v1 - v0) * frac;
        for (int b = 0; b < B_; ++b) stats[1 + b] = smed[b];
    }
}

// ---------------------------------------------------------------------------
// Flash-style complex spectral attention.
// Block = 128 threads (4 waves) handles one (b, 16-row f-tile).
// Pass 1: A fragments (x_fft panel, 128 VGPRs) preloaded ONCE into registers,
//         reused across all 129 g-tiles; scores via 4 fp32 WMMA per K-step.
// Softmax in 132KB dynamic LDS. Pass 2: context = P @ V, c split across waves.
// ---------------------------------------------------------------------------
__global__ __launch_bounds__(128) void attn_kernel(
    const float* __restrict__ Xre, const float* __restrict__ Xim,
    const float* __restrict__ Kre, const float* __restrict__ Kim,
    const float* __restrict__ Vre, const float* __restrict__ Vim,
    float* __restrict__ CtxRe, float* __restrict__ CtxIm)
{
    extern __shared__ float P[];                      // [16][GP]
    const int wave = threadIdx.x >> 5;
    const int lane = threadIdx.x & 31;
    const int lm = lane & 15, lh = lane >> 4;
    const int b  = blockIdx.x / NGT;
    const int f0 = (blockIdx.x % NGT) * 16;
    const float scale = 0.08838834764831845f;         // 1/sqrt(128)

    const int fA = min(f0 + lm, F_ - 1);
    const float* xrRow = Xre + ((size_t)b * F_ + fA) * C_;
    const float* xiRow = Xim + ((size_t)b * F_ + fA) * C_;

    // ----- preload A fragments for the full K=128 reduction (64 v2f regs) -----
    v2f arf[32], aif[32];
#pragma unroll
    for (int kq = 0; kq < 32; ++kq) {
        const int ka = kq * 4 + 2 * lh;
        arf[kq] = v2f{ xrRow[ka], xrRow[ka + 1] };
        aif[kq] = v2f{ xiRow[ka], xiRow[ka + 1] };
    }

    // ----- pass 1: score panel -----
    for (int gt = wave; gt < NGT; gt += 4) {
        const int g0 = gt * 16;
        const int gB = min(g0 + lm, F_ - 1);
        const float* krRow = Kre + ((size_t)b * F_ + gB) * C_;
        const float* kiRow = Kim + ((size_t)b * F_ + gB) * C_;
        v8f sre = {}, sim = {};
#pragma unroll
        for (int kq = 0; kq < 32; ++kq) {
            const int ka = kq * 4 + 2 * lh;
            v2f br = { krRow[ka], krRow[ka + 1] };
            v2f bi = { kiRow[ka], kiRow[ka + 1] };
            sre = wmma_acc(arf[kq],  br, sre);
            sre = wmma_acc(-aif[kq], bi, sre);        // re -= ai*bi
            sim = wmma_acc(arf[kq],  bi, sim);
            sim = wmma_acc(aif[kq],  br, sim);
        }
#pragma unroll
        for (int r = 0; r < 8; ++r) {
            const int row = r + 8 * lh;
            const int g = g0 + lm;
            const float v = sqrtf(sre[r] * sre[r] + sim[r] * sim[r]) * scale;
            P[row * GP + g] = (g < F_) ? v : -INFINITY;
        }
    }
    __syncthreads();

    // ----- softmax over g (per row), zero the tail -----
    for (int row = wave; row < 16; row += 4) {
        float m = -INFINITY;
        for (int g = lane; g < F_; g += 32) m = fmaxf(m, P[row * GP + g]);
        for (int off = 16; off; off >>= 1) m = fmaxf(m, __shfl_xor(m, off));
        float s = 0.0f;
        for (int g = lane; g < F_; g += 32) {
            const float e = __expf(P[row * GP + g] - m);
            P[row * GP + g] = e;
            s += e;
        }
        for (int off = 16; off; off >>= 1) s += __shfl_xor(s, off);
        const float inv = 1.0f / s;
        for (int g = lane; g < GP; g += 32)
            P[row * GP + g] = (g < F_) ? P[row * GP + g] * inv : 0.0f;
    }
    __syncthreads();

    // ----- pass 2: context = P @ V (wave w owns channels [32w, 32w+32)) -----
    const int c0a = wave * 32, c0b = c0a + 16;
    v8f ore0 = {}, oim0 = {}, ore1 = {}, oim1 = {};
    for (int gt = 0; gt < NGT; ++gt) {
        const int g0 = gt * 16;
        const int gpf = min(g0 + 16, F_ - 1);
        __builtin_prefetch(Vre + ((size_t)b * F_ + gpf) * C_ + c0a, 0, 1);
        __builtin_prefetch(Vim + ((size_t)b * F_ + gpf) * C_ + c0a, 0, 1);
#pragma unroll
        for (int kk = 0; kk < 16; kk += 4) {
            const int ga = g0 + kk + 2 * lh;
            v2f a = { P[lm * GP + ga], P[lm * GP + ga + 1] };
            const int g1i = min(ga, F_ - 1), g2i = min(ga + 1, F_ - 1);
            const float* vr1 = Vre + ((size_t)b * F_ + g1i) * C_;
            const float* vr2 = Vre + ((size_t)b * F_ + g2i) * C_;
            const float* vi1 = Vim + ((size_t)b * F_ + g1i) * C_;
            const float* vi2 = Vim + ((size_t)b * F_ + g2i) * C_;
            v2f br0 = { vr1[c0a + lm], vr2[c0a + lm] };
            v2f bi0 = { vi1[c0a + lm], vi2[c0a + lm] };
            v2f br1 = { vr1[c0b + lm], vr2[c0b + lm] };
            v2f bi1 = { vi1[c0b + lm], vi2[c0b + lm] };
            ore0 = wmma_acc(a, br0, ore0);
            oim0 = wmma_acc(a, bi0, oim0);
            ore1 = wmma_acc(a, br1, ore1);
            oim1 = wmma_acc(a, bi1, oim1);
        }
    }
#pragma unroll
    for (int r = 0; r < 8; ++r) {
        const int f = f0 + r + 8 * lh;
        if (f < F_) {
            const size_t o = ((size_t)b * F_ + f) * C_;
            CtxRe[o + c0a + lm] = ore0[r];
            CtxIm[o + c0a + lm] = oim0[r];
            CtxRe[o + c0b + lm] = ore1[r];
            CtxIm[o + c0b + lm] = oim1[r];
        }
    }
}

// ---------------------------------------------------------------------------
// context += where(normalized > threshold, x_fft, 0) * (w_high[:,0] + i*w_high[:,1])
// ---------------------------------------------------------------------------
__global__ __launch_bounds__(128) void combine_kernel(
    const float* __restrict__ Xre, const float* __restrict__ Xim,
    const float* __restrict__ energy, const float* __restrict__ stats,
    const float* __restrict__ w_high,
    float* __restrict__ CtxRe, float* __restrict__ CtxIm)
{
    const int idx = blockIdx.x;                       // b*F + f
    const int b = idx / F_;
    const float norm = energy[idx] / (stats[1 + b] + 1e-6f);
    if (!(norm > stats[0])) return;
    const int c = threadIdx.x;
    const float whr = w_high[2 * c], whi = w_high[2 * c + 1];
    const size_t o = (size_t)idx * C_ + c;
    const float xr = Xre[o], xi = Xim[o];
    CtxRe[o] += xr * whr - xi * whi;
    CtxIm[o] += xr * whi + xi * whr;
}

// ---------------------------------------------------------------------------
extern "C" void kernel_launch(void* const* d_in, const int* in_sizes, int n_in,
                              void* d_out, int out_size, void* d_ws, size_t ws_size,
                              hipStream_t stream)
{
    (void)in_sizes; (void)n_in; (void)out_size; (void)ws_size;
    const float* x_in   = (const float*)d_in[0];
    const float* W_K    = (const float*)d_in[1];
    const float* b_K    = (const float*)d_in[2];
    const float* W_V    = (const float*)d_in[3];
    const float* b_V    = (const float*)d_in[4];
    const float* w_high = (const float*)d_in[5];
    const float* thr    = (const float*)d_in[6];
    float* out = (float*)d_out;

    char* ws = (char*)d_ws;
    size_t off = 0;
    auto alloc = [&](size_t nf) { float* p = (float*)(ws + off); off += nf * sizeof(float); return p; };
    float* xKt = alloc((size_t)B_ * C_ * N_);         // 16 MB (later reused as CtxRe)
    float* xVt = alloc((size_t)B_ * C_ * N_);         // 16 MB (later reused as CtxIm)
    float* Xre = alloc((size_t)B_ * F_ * C_);
    float* Xim = alloc((size_t)B_ * F_ * C_);
    float* Kre = alloc((size_t)B_ * F_ * C_);
    float* Kim = alloc((size_t)B_ * F_ * C_);
    float* Vre = alloc((size_t)B_ * F_ * C_);
    float* Vim = alloc((size_t)B_ * F_ * C_);
    float* energy = alloc((size_t)B_ * F_);
    float* stats  = alloc(16);
    float* CtxRe = xKt;                               // xKt/xVt dead after their FFTs
    float* CtxIm = xVt;

    proj_kernel<<<2048, 256, 0, stream>>>(x_in, W_K, b_K, W_V, b_V, xKt, xVt);
    fft_fwd_kernel<<<B_ * C_, 256, 0, stream>>>(x_in, Xre, Xim);
    fft_fwd_kernel<<<B_ * C_, 256, 0, stream>>>(xKt, Kre, Kim);
    fft_fwd_kernel<<<B_ * C_, 256, 0, stream>>>(xVt, Vre, Vim);
    energy_kernel<<<B_ * F_, 32, 0, stream>>>(Xre, Xim, energy);
    medquant_kernel<<<1, 256, 0, stream>>>(energy, thr, stats);
    attn_kernel<<<B_ * NGT, 128, 16 * GP * sizeof(float), stream>>>(
        Xre, Xim, Kre, Kim, Vre, Vim, CtxRe, CtxIm);
    combine_kernel<<<B_ * F_, 128, 0, stream>>>(Xre, Xim, energy, stats, w_high,
                                                CtxRe, CtxIm);
    fft_inv_kernel<<<B_ * C_, 256, 0, stream>>>(CtxRe, CtxIm, out);
}